// GPUTrieTokenizer_36077725286430
// MI455X (gfx1250) — compile-verified
//
#include <hip/hip_runtime.h>
#include <hip/hip_bf16.h>

// GPUTrieTokenizer for MI455X (gfx1250, wave32).
// Phase 1: parallel longest-match trie walk per (row, position). 64MB children
//          table is L2-resident (192MB L2); root row + sequence window staged
//          into LDS via the CDNA5 async copy engine
//          (global_load_async_to_lds_b128 / ASYNCcnt / s_wait_asynccnt).
// Phase 2: per-row greedy scan. Because max pattern length (6) < chunk width
//          (8), each 8-position chunk's jump-through (exit, token-count) is
//          precomputed in parallel; the serial greedy chain shrinks from 2048
//          dependent LDS loads to 256 chunk hops, then chunks replay in
//          parallel to compact tokens, followed by coalesced output stores.

#define TRIE_B 1024
#define TRIE_L 2048
#define TRIE_MAXPAT 8
#define CHUNK 8
#define NCHUNK (TRIE_L / CHUNK)   // 256

__device__ __forceinline__ void async_g2l_b128(const void* gsrc, void* ldst) {
    // VDST of the async op is the workgroup-relative LDS byte offset, which is
    // exactly the low 32 bits of a generic __shared__ pointer on gfx1250.
    unsigned int loff = (unsigned int)(unsigned long long)ldst;
    unsigned long long ga = (unsigned long long)gsrc;
    asm volatile("global_load_async_to_lds_b128 %0, %1, off"
                 :: "v"(loff), "v"(ga) : "memory");
}

__device__ __forceinline__ void wait_async_zero() {
    asm volatile("s_wait_asynccnt 0" ::: "memory");
}

// ---------------------------------------------------------------------------
// Phase 1: one 256-thread block handles 256 consecutive positions of one row.
// Packs (step_len << 24) | step_tok per position into ws.
// ---------------------------------------------------------------------------
__global__ void trie_match_kernel(const int* __restrict__ seqs,
                                  const int* __restrict__ lens,
                                  const int* __restrict__ children,
                                  const int* __restrict__ output,
                                  const int* __restrict__ length,
                                  unsigned int* __restrict__ packed) {
    __shared__ __align__(16) int s_seq[272];   // 256 positions + 8 lookahead
    __shared__ __align__(16) int s_root[256];  // children row of trie root

    const int b  = blockIdx.y;
    const int i0 = blockIdx.x * 256;
    const int t  = threadIdx.x;

    // Async-stage seq window (66 x b128) and root children row (64 x b128).
    if (t < 66) {
        long long gidx = (long long)b * TRIE_L + i0 + t * 4;
        const long long maxIdx = (long long)TRIE_B * TRIE_L - 4;
        if (gidx > maxIdx) gidx = maxIdx;   // clamped data is never consumed
        async_g2l_b128(seqs + gidx, &s_seq[t * 4]);
    } else if (t < 130) {
        const int k = (t - 66) * 4;
        async_g2l_b128(children + k, &s_root[k]);
    }
    wait_async_zero();
    __syncthreads();

    const int myLen = lens[b];
    const int i = i0 + t;

    int node = 0;
    int best_tok = -1;
    int best_len = 0;
#pragma unroll
    for (int j = 0; j < TRIE_MAXPAT; ++j) {
        if (i + j >= myLen) break;                 // in_range gate (len <= L)
        const int byte_j = s_seq[t + j];
        const int child = (j == 0) ? s_root[byte_j]
                                   : children[node * 256 + byte_j];
        if (child < 0) break;                      // valid &= (child != -1)
        node = child;
        const int tok = output[node];
        if (tok >= 0) {
            best_tok = tok;
            best_len = length[node];
        }
    }

    const bool use_match = (best_tok >= 0) && (best_len > 1);
    const unsigned int sl = use_match ? (unsigned int)best_len : 1u;
    const unsigned int st = use_match ? (unsigned int)best_tok : (unsigned int)s_seq[t];
    packed[(size_t)b * TRIE_L + i] = (sl << 24) | (st & 0x00FFFFFFu);
}

// ---------------------------------------------------------------------------
// Phase 2: one block (256 threads) per row.
//   1) async copy packed row -> LDS
//   2) per-chunk backward DP: s_jump[p] = (exit_pos << 4) | tokens_in_chunk
//      (max jump 6 < 8  =>  exit always lands in the next chunk)
//   3) lane 0: 256-hop serial chunk chain, records (entry, token_base)/chunk;
//      final partial chunk walked finely with the p < len guard
//   4) parallel chunk replay into compacted LDS arrays (re-checks p < len)
//   5) cooperative zero-padded coalesced stores
// ---------------------------------------------------------------------------
__global__ void greedy_scan_kernel(const unsigned int* __restrict__ packed,
                                   const int* __restrict__ lens,
                                   float* __restrict__ token_ids,
                                   float* __restrict__ char_pos,
                                   float* __restrict__ char_len,
                                   float* __restrict__ num_tok) {
    __shared__ __align__(16) unsigned int s_pack[TRIE_L];   // 8KB packed matches
    __shared__ unsigned short s_jump[TRIE_L];               // 4KB (exit<<4)|cnt
    __shared__ unsigned int s_entry[NCHUNK];                // 1KB entry|(tb<<16)
    __shared__ unsigned int s_tok[TRIE_L];                  // 8KB compacted tokens
    __shared__ unsigned int s_pl[TRIE_L];                   // 8KB pos|(len<<16)
    __shared__ int s_total;

    const int b = blockIdx.x;
    const int t = threadIdx.x;

    // (1) 2048 u32 = 512 b128 async transfers; 2 per thread.
    for (int k = t; k < TRIE_L / 4; k += blockDim.x) {
        async_g2l_b128(packed + (size_t)b * TRIE_L + (size_t)k * 4, &s_pack[k * 4]);
    }
    s_entry[t] = 0xFFFFFFFFu;
    wait_async_zero();
    __syncthreads();

    // (2) intra-chunk jump-through DP (backward; no cross-thread deps).
    {
        const int base = t * CHUNK;
#pragma unroll
        for (int e = CHUNK - 1; e >= 0; --e) {
            const int p   = base + e;
            const int sl  = (int)(s_pack[p] >> 24);
            const int tgt = p + sl;                 // sl in [1,6] => tgt <= base+13
            unsigned short v;
            if (tgt < base + CHUNK) {
                const unsigned short w = s_jump[tgt];
                v = (unsigned short)((w & 0xFFF0u) | ((w & 0xFu) + 1u));
            } else {
                v = (unsigned short)(((unsigned)tgt << 4) | 1u);
            }
            s_jump[p] = v;
        }
    }
    __syncthreads();

    // (3) serial chunk chain (<=256 dependent LDS hops).
    if (t == 0) {
        const int myLen = lens[b];
        int nxt = 0, tb = 0;
        while (nxt < myLen) {
            const int c = nxt >> 3;
            s_entry[c] = (unsigned int)nxt | ((unsigned int)tb << 16);
            if ((c + 1) * CHUNK <= myLen) {         // full chunk: use jump table
                const unsigned short w = s_jump[nxt];
                nxt = (int)(w >> 4);
                tb += (int)(w & 0xFu);
            } else {                                // final partial chunk
                while (nxt < myLen) {
                    nxt += (int)(s_pack[nxt] >> 24);
                    ++tb;
                }
            }
        }
        s_total = tb;
    }
    __syncthreads();

    // (4) parallel replay: thread t replays chunk t (if entered).
    {
        const int myLen = lens[b];
        const unsigned int e = s_entry[t];
        if (e != 0xFFFFFFFFu) {
            int p  = (int)(e & 0xFFFFu);
            int tb = (int)(e >> 16);
            const int lim = min((t + 1) * CHUNK, myLen);
            while (p < lim) {
                const unsigned int pk = s_pack[p];
                const int sl = (int)(pk >> 24);
                s_tok[tb] = pk & 0x00FFFFFFu;
                s_pl[tb]  = (unsigned int)p | ((unsigned int)sl << 16);
                ++tb;
                p += sl;
            }
        }
    }
    __syncthreads();

    // (5) zero-padded coalesced stores (values < 2^24: exact in f32).
    const int cnt = s_total;
    const size_t base = (size_t)b * TRIE_L;
    for (int k = t; k < TRIE_L; k += blockDim.x) {
        float tok = 0.0f, cp = 0.0f, cl = 0.0f;
        if (k < cnt) {
            tok = (float)s_tok[k];
            const unsigned int pl = s_pl[k];
            cp = (float)(pl & 0xFFFFu);
            cl = (float)(pl >> 16);
        }
        token_ids[base + k] = tok;
        char_pos[base + k]  = cp;
        char_len[base + k]  = cl;
    }
    if (t == 0) num_tok[b] = (float)cnt;
}

extern "C" void kernel_launch(void* const* d_in, const int* in_sizes, int n_in,
                              void* d_out, int out_size, void* d_ws, size_t ws_size,
                              hipStream_t stream) {
    (void)in_sizes; (void)n_in; (void)out_size; (void)ws_size;

    const int* seqs     = (const int*)d_in[0];  // (B, L) int32
    const int* lens     = (const int*)d_in[1];  // (B,)   int32
    const int* children = (const int*)d_in[2];  // (65536, 256) int32
    const int* output   = (const int*)d_in[3];  // (65536,) int32
    const int* length   = (const int*)d_in[4];  // (65536,) int32

    unsigned int* packed = (unsigned int*)d_ws;  // B*L u32 = 8MB scratch

    float* out       = (float*)d_out;
    float* token_ids = out;                                    // B*L
    float* char_pos  = out + (size_t)TRIE_B * TRIE_L;          // B*L
    float* char_len  = out + (size_t)2 * TRIE_B * TRIE_L;      // B*L
    float* num_tok   = out + (size_t)3 * TRIE_B * TRIE_L;      // B

    dim3 grid1(TRIE_L / 256, TRIE_B);
    trie_match_kernel<<<grid1, 256, 0, stream>>>(seqs, lens, children, output,
                                                 length, packed);

    greedy_scan_kernel<<<TRIE_B, 256, 0, stream>>>(packed, lens, token_ids,
                                                   char_pos, char_len, num_tok);
}